// ExplicitREN_89300960019170
// MI455X (gfx1250) — compile-verified
//
#include <hip/hip_runtime.h>
#include <math.h>

typedef __attribute__((ext_vector_type(2))) float        v2f;
typedef __attribute__((ext_vector_type(8))) float        v8f;
typedef __attribute__((ext_vector_type(4))) unsigned int u32x4;
typedef __attribute__((ext_vector_type(8))) int          i32x8;
typedef __attribute__((ext_vector_type(4))) int          i32x4;

#define BATCH   8192
#define IN_DIM  128
#define HID     1024
#define OUT_DIM 128
#define KB      16            // hidden block (WMMA tile M/N)
#define NB      (HID / KB)    // 64 hidden blocks
#define BT      32            // batch tile per workgroup
#define NW      (BT / 16)     // waves per workgroup
#define NTHR    (NW * 32)

// fp32 WMMA: D(16x16) = A(16x4) * B(4x16) + C. Layouts per ISA 7.12.2:
// A: VGPR0={K0|K2}, VGPR1={K1|K3} -> lane holds float2 at K = k0 + 2*(lane>=16)
// B: assumed analogous striping, N = lane&15
// C/D: VGPR r -> lanes0-15: M=r, N=lane ; lanes16-31: M=r+8, N=lane-16
__device__ __forceinline__ v8f wmma_f32_4(v2f a, v2f b, v8f c) {
    return __builtin_amdgcn_wmma_f32_16x16x4_f32(false, a, false, b, (short)0, c,
                                                 false, false);
}

// Issue a TDM load of a 2D tile of Bs into LDS (packed rows):
//   tile = Bs[row0 .. row0+15][0 .. width-1], tensor row stride = HID elements.
// D# bitfields per CDNA5 ISA ch.8 (group0: flags/lds/global addr, group1: dims).
__device__ __forceinline__ void tdm_load_bs_rows(const float* gsrc,
                                                 unsigned int lds_byte_addr,
                                                 int width_elems) {
    const unsigned long long ga = (unsigned long long)(size_t)gsrc;

    u32x4 g0;
    g0.x = 1u;                                   // count=1 (valid), user descriptor
    g0.y = lds_byte_addr;                        // lds_addr (bytes)
    g0.z = (unsigned int)ga;                     // global_addr[31:0]
    g0.w = (unsigned int)((ga >> 32) & 0x01FFFFFFu) | (2u << 30); // addr[56:32]|type=2

    i32x8 g1;
    g1[0] = (int)(2u << 16);                     // workgroup_mask=0, data_size=2 (4B)
    g1[1] = (int)((unsigned)(HID & 0xFFFF) << 16);        // tensor_dim0[15:0]
    g1[2] = (int)((HID >> 16) | ((unsigned)(HID & 0xFFFF) << 16)); // td0 hi | td1 lo
    g1[3] = (int)((HID >> 16) | ((unsigned)width_elems << 16));    // td1 hi | tile_dim0
    g1[4] = KB;                                  // tile_dim1 = 16 rows, tile_dim2 = 0
    g1[5] = HID;                                 // tensor_dim0_stride[31:0]
    g1[6] = 0;                                   // stride hi | tensor_dim1_stride lo
    g1[7] = 0;

    i32x4 gz4 = {0, 0, 0, 0};                    // groups 2/3 unused (2D tensor)
    i32x8 gz8 = {0, 0, 0, 0, 0, 0, 0, 0};        // trailing group (clang-23 6-arg form)
    __builtin_amdgcn_tensor_load_to_lds(g0, g1, gz4, gz4, gz8, 0);
}

__global__ __launch_bounds__(NTHR, 1)
void ren_fused_kernel(const float* __restrict__ U,     // [BATCH][IN_DIM]
                      const float* __restrict__ Bw,    // [HID][IN_DIM]
                      const float* __restrict__ BsF,   // [HID][HID]
                      const float* __restrict__ Dsw,   // [OUT_DIM][HID]
                      const float* __restrict__ Dw,    // [OUT_DIM][IN_DIM]
                      float* __restrict__ Y)           // [BATCH][OUT_DIM]
{
    __shared__ float sS[HID][BT];               // 128 KB: state, resident all kernel
    __shared__ float uT[IN_DIM][BT];            // 16 KB: u tile, transposed
    __shared__ float bsTile[2][KB * HID];       // 2x64 KB: TDM double buffer (packed)
    __shared__ float accS[NW][KB][KB + 1];      // per-wave C spill for seq. phase

    const int tid   = threadIdx.x;
    const int lane  = tid & 31;
    const int wave  = tid >> 5;
    const int half  = lane >> 4;
    const int l16   = lane & 15;
    const int bBase = blockIdx.x * BT;
    const int col   = wave * 16 + l16;          // batch column in tile

    // ---- stage u tile transposed: uT[k][b] = U[bBase+b][k] ----
    for (int idx = tid; idx < BT * IN_DIM; idx += NTHR) {
        const int b = idx / IN_DIM;
        const int k = idx - b * IN_DIM;
        uT[k][b] = U[(size_t)(bBase + b) * IN_DIM + k];
    }

    // ---- TDM prologue: DMA block 0's Bs rows (width 16) into buffer 0 ----
    if (wave == 0) {
        tdm_load_bs_rows(BsF, (unsigned int)(size_t)&bsTile[0][0], KB);
    }
    __syncthreads();

    // ================= blocked forward substitution =================
    for (int ib = 0; ib < NB; ++ib) {
        const int i0    = ib * KB;
        const int buf   = ib & 1;
        const int width = i0 + KB;              // packed LDS tile row pitch

        // issue DMA for next block into the other buffer, then wait for current
        if (wave == 0) {
            if (ib + 1 < NB) {
                tdm_load_bs_rows(&BsF[(size_t)(i0 + KB) * HID],
                                 (unsigned int)(size_t)&bsTile[buf ^ 1][0],
                                 width + KB);
                __builtin_amdgcn_s_wait_tensorcnt(1);  // oldest (current) done
            } else {
                __builtin_amdgcn_s_wait_tensorcnt(0);
            }
        }
        __syncthreads();                         // current tile visible to all waves

        const float* atile = &bsTile[buf][(size_t)l16 * width];

        v8f acc = {};   // C tile: 16 hidden rows x 16 batch cols, fp32

        // --- Bu contribution: A = Bw[i0..i0+15][:] (global), B = uT (LDS) ---
        {
            const float* arow = &Bw[(size_t)(i0 + l16) * IN_DIM];
            for (int k0 = 0; k0 < IN_DIM; k0 += 4) {
                v2f a = *(const v2f*)(arow + k0 + half * 2);
                v2f b;
                b.x = uT[k0 + half * 2 + 0][col];
                b.y = uT[k0 + half * 2 + 1][col];
                acc = wmma_f32_4(a, b, acc);
            }
        }
        // --- inter-block recurrence: A from TDM-staged LDS tile, B = s (LDS) ---
        for (int jb = 0; jb < ib; ++jb) {
            const int j0 = jb * KB;
#pragma unroll
            for (int kk = 0; kk < KB; kk += 4) {
                v2f a = *(const v2f*)(atile + j0 + kk + half * 2);
                v2f b;
                b.x = sS[j0 + kk + half * 2 + 0][col];
                b.y = sS[j0 + kk + half * 2 + 1][col];
                acc = wmma_f32_4(a, b, acc);
            }
        }

        // spill C so each lane owns one full batch column
#pragma unroll
        for (int r = 0; r < 8; ++r)
            accS[wave][half * 8 + r][l16] = acc[r];
        __syncthreads();

        // --- sequential 16-step tanh triangle (lanes 0-15 own one batch col) ---
        if (lane < 16) {
            const float* dtri = &bsTile[buf][i0];   // diag block, row pitch = width
            float a[KB];
#pragma unroll
            for (int m = 0; m < KB; ++m) a[m] = accS[wave][m][l16];
#pragma unroll
            for (int k = 0; k < KB; ++k) {
                const float sv = tanhf(a[k]);
                sS[i0 + k][wave * 16 + l16] = sv;
#pragma unroll
                for (int m = k + 1; m < KB; ++m)
                    a[m] += dtri[(size_t)m * width + k] * sv;   // strict lower
            }
        }
        __syncthreads();   // sS visible; buf safe to recycle next-next iter
    }

    // warm the output weights (gfx1250 global_prefetch path)
    __builtin_prefetch(&Dsw[(size_t)tid * HID], 0, 1);
    __builtin_prefetch(&Dsw[(size_t)(tid + NTHR) * HID], 0, 1);
    __builtin_prefetch(&Dw[(size_t)(tid & 127) * IN_DIM], 0, 1);

    // ================= output: y = s @ Ds^T + u @ D^T =================
    for (int ob = 0; ob < OUT_DIM / KB; ++ob) {
        const int o0 = ob * KB;
        v8f acc = {};
        {
            const float* arow = &Dsw[(size_t)(o0 + l16) * HID];
#pragma unroll 4
            for (int k0 = 0; k0 < HID; k0 += 4) {
                v2f a = *(const v2f*)(arow + k0 + half * 2);
                v2f b;
                b.x = sS[k0 + half * 2 + 0][col];
                b.y = sS[k0 + half * 2 + 1][col];
                acc = wmma_f32_4(a, b, acc);
            }
        }
        {
            const float* drow = &Dw[(size_t)(o0 + l16) * IN_DIM];
#pragma unroll 4
            for (int k0 = 0; k0 < IN_DIM; k0 += 4) {
                v2f a = *(const v2f*)(drow + k0 + half * 2);
                v2f b;
                b.x = uT[k0 + half * 2 + 0][col];
                b.y = uT[k0 + half * 2 + 1][col];
                acc = wmma_f32_4(a, b, acc);
            }
        }
        // C layout: lane holds out-cols o0 + half*8 + 0..7 for batch row (bBase+col)
        float* op = &Y[(size_t)(bBase + col) * OUT_DIM + o0 + half * 8];
        *(float4*)(op + 0) = make_float4(acc[0], acc[1], acc[2], acc[3]);
        *(float4*)(op + 4) = make_float4(acc[4], acc[5], acc[6], acc[7]);
    }
}

extern "C" void kernel_launch(void* const* d_in, const int* in_sizes, int n_in,
                              void* d_out, int out_size, void* d_ws, size_t ws_size,
                              hipStream_t stream) {
    (void)in_sizes; (void)n_in; (void)d_ws; (void)ws_size; (void)out_size;
    const float* u   = (const float*)d_in[0];
    const float* Bw  = (const float*)d_in[1];
    const float* Bs  = (const float*)d_in[2];
    const float* Dsw = (const float*)d_in[3];
    const float* Dw  = (const float*)d_in[4];
    float* out = (float*)d_out;

    hipLaunchKernelGGL(ren_fused_kernel, dim3(BATCH / BT), dim3(NTHR), 0, stream,
                       u, Bw, Bs, Dsw, Dw, out);
}